// TGCN_11836929868497
// MI455X (gfx1250) — compile-verified
//
#include <hip/hip_runtime.h>
#include <hip/hip_bf16.h>
#include <math.h>

// ---------------------------------------------------------------------------
// TGCN cell for MI455X (gfx1250, wave32, WMMA).
//
// Algebraic refactor: GCNConv aggregation is linear, so
//   Z_k = Ahat @ (X @ W_k) + b_k  ==  (Ahat @ X) @ W_k + b_k
// -> aggregate X ONCE over the 1.6M edges, then three dense GEMMs.
// GEMMs use V_WMMA_F32_16X16X4_F32 (exact fp32 matrix math), 32 rows/block
// so each weight fetch feeds two WMMA row-tiles. The candidate-gate GEMM
// fuses the final GRU combine (Ht = U*H + (1-U)*tanh(.)) in its epilogue.
// ---------------------------------------------------------------------------

#define N_NODES 100000
#define N_EDGES 1600000
#define C 128          // IN_C == OUT_C
#define NC (N_NODES * C)

typedef __attribute__((ext_vector_type(2))) float v2f;
typedef __attribute__((ext_vector_type(8))) float v8f;

__device__ __forceinline__ v8f wmma_f32x4(v2f a, v2f b, v8f c) {
  // (neg_a, A, neg_b, B, c_mod, C, reuse_a, reuse_b)
  return __builtin_amdgcn_wmma_f32_16x16x4_f32(false, a, false, b, (short)0, c,
                                               false, false);
}

// --------------------------- degree / dinv ---------------------------------
__global__ __launch_bounds__(256) void k_zero_i32(int* p, int n) {
  int i = blockIdx.x * 256 + threadIdx.x;
  if (i < n) p[i] = 0;
}

__global__ __launch_bounds__(256) void k_count_deg(const int* __restrict__ dst,
                                                   int* __restrict__ cnt) {
  int e = blockIdx.x * 256 + threadIdx.x;  // grid covers exactly N_EDGES
  atomicAdd(&cnt[dst[e]], 1);
}

__global__ __launch_bounds__(256) void k_make_dinv(const int* __restrict__ cnt,
                                                   float* __restrict__ dinv,
                                                   int n) {
  int i = blockIdx.x * 256 + threadIdx.x;
  if (i < n) {
    float deg = (float)cnt[i] + 1.0f;  // +1 self loop
    dinv[i] = rsqrtf(deg);
  }
}

// ----------------------- aggregation: Agg = Ahat @ X -----------------------
__global__ __launch_bounds__(256) void k_self_init(const float* __restrict__ X,
                                                   const float* __restrict__ dinv,
                                                   float* __restrict__ Agg) {
  int idx = blockIdx.x * 256 + threadIdx.x;  // over N*C, exact
  float di = dinv[idx >> 7];
  Agg[idx] = di * di * X[idx];
}

__global__ __launch_bounds__(256) void k_edge_agg(const int* __restrict__ src,
                                                  const int* __restrict__ dst,
                                                  const float* __restrict__ X,
                                                  const float* __restrict__ dinv,
                                                  float* __restrict__ Agg) {
  // 32 lanes per edge, 4 channels per lane (float4 gather, 4 float atomics).
  int e  = blockIdx.x * 8 + (threadIdx.x >> 5);  // grid = E/8 blocks, exact
  int ch = (threadIdx.x & 31) * 4;
  int s = src[e], d = dst[e];
  float coef = dinv[s] * dinv[d];
  const float4 x = *(const float4*)&X[(size_t)s * C + ch];
  float* p = &Agg[(size_t)d * C + ch];
  __hip_atomic_fetch_add(&p[0], coef * x.x, __ATOMIC_RELAXED, __HIP_MEMORY_SCOPE_AGENT);
  __hip_atomic_fetch_add(&p[1], coef * x.y, __ATOMIC_RELAXED, __HIP_MEMORY_SCOPE_AGENT);
  __hip_atomic_fetch_add(&p[2], coef * x.z, __ATOMIC_RELAXED, __HIP_MEMORY_SCOPE_AGENT);
  __hip_atomic_fetch_add(&p[3], coef * x.w, __ATOMIC_RELAXED, __HIP_MEMORY_SCOPE_AGENT);
}

// --------------------- GEMM 1: Z = Agg @ W(128x128) + b --------------------
// block = 256 threads (8 waves); handles 32 rows x 128 cols (two 16-row
// tiles share each B register pair). wave w owns columns [16w, 16w+16).
__global__ __launch_bounds__(256) void k_gemm_xw(const float* __restrict__ A,
                                                 const float* __restrict__ W,
                                                 const float* __restrict__ bias,
                                                 float* __restrict__ Z) {
  __shared__ float sA[32 * C];  // 16 KB
  const int rowBase = blockIdx.x * 32;
  for (int i = threadIdx.x; i < 32 * C; i += 256) {
    int r = i >> 7, k = i & (C - 1);
    sA[i] = A[(size_t)(rowBase + r) * C + k];
  }
  __syncthreads();

  const int lane = threadIdx.x & 31;
  const int wave = threadIdx.x >> 5;
  const int hi = lane >> 4;   // selects K pair {0,1} vs {2,3}
  const int lo = lane & 15;   // A row / B col within tile
  const int col = wave * 16 + lo;

  v8f acc0 = {}, acc1 = {};
#pragma unroll 4
  for (int k0 = 0; k0 < C; k0 += 4) {
    int kb = k0 + 2 * hi;
    v2f b;
    b.x = W[kb * C + col];
    b.y = W[(kb + 1) * C + col];
    v2f a0 = *(const v2f*)&sA[lo * C + kb];
    v2f a1 = *(const v2f*)&sA[(16 + lo) * C + kb];
    acc0 = wmma_f32x4(a0, b, acc0);
    acc1 = wmma_f32x4(a1, b, acc1);
  }

  float bv = bias[col];
#pragma unroll
  for (int j = 0; j < 8; ++j) {
    int r = j + 8 * hi;  // C/D layout: lanes>=16 hold rows M+8
    Z[(size_t)(rowBase + r) * C + col] = acc0[j] + bv;
    Z[(size_t)(rowBase + 16 + r) * C + col] = acc1[j] + bv;
  }
}

// ------- GEMM 2: Out = act( [A1 | A2(*Rm)] @ L(256x128) + b ) --------------
// act: 0 = sigmoid, 1 = tanh. Rm==nullptr -> A2 used as-is.
// If Ufin != nullptr (final mode): Out = Ufin*Hfin + (1-Ufin)*tanh(x).
__global__ __launch_bounds__(256) void k_gemm_gate(const float* __restrict__ A1,
                                                   const float* __restrict__ A2,
                                                   const float* __restrict__ Rm,
                                                   const float* __restrict__ L,
                                                   const float* __restrict__ bias,
                                                   float* __restrict__ Out,
                                                   int act,
                                                   const float* __restrict__ Ufin,
                                                   const float* __restrict__ Hfin) {
  __shared__ float sA[32 * 2 * C];  // 32 rows x 256 (concat) = 32 KB
  const int rowBase = blockIdx.x * 32;
  for (int i = threadIdx.x; i < 32 * 2 * C; i += 256) {
    int r = i >> 8, k = i & (2 * C - 1);
    int gr = rowBase + r;
    float v;
    if (k < C) {
      v = A1[(size_t)gr * C + k];
    } else {
      v = A2[(size_t)gr * C + (k - C)];
      if (Rm) v *= Rm[(size_t)gr * C + (k - C)];
    }
    sA[i] = v;
  }
  __syncthreads();

  const int lane = threadIdx.x & 31;
  const int wave = threadIdx.x >> 5;
  const int hi = lane >> 4;
  const int lo = lane & 15;
  const int col = wave * 16 + lo;

  v8f acc0 = {}, acc1 = {};
#pragma unroll 4
  for (int k0 = 0; k0 < 2 * C; k0 += 4) {
    int kb = k0 + 2 * hi;
    v2f b;
    b.x = L[kb * C + col];
    b.y = L[(kb + 1) * C + col];
    v2f a0 = *(const v2f*)&sA[lo * (2 * C) + kb];
    v2f a1 = *(const v2f*)&sA[(16 + lo) * (2 * C) + kb];
    acc0 = wmma_f32x4(a0, b, acc0);
    acc1 = wmma_f32x4(a1, b, acc1);
  }

  float bv = bias[col];
#pragma unroll
  for (int j = 0; j < 8; ++j) {
    int r = j + 8 * hi;
#pragma unroll
    for (int t = 0; t < 2; ++t) {
      size_t idx = (size_t)(rowBase + 16 * t + r) * C + col;
      float x = (t == 0 ? acc0[j] : acc1[j]) + bv;
      if (Ufin) {
        // fused GRU combine: Ht = U*H + (1-U)*tanh(x)
        float u = Ufin[idx];
        float h = Hfin[idx];
        Out[idx] = u * h + (1.0f - u) * tanhf(x);
      } else {
        x = (act == 0) ? (1.0f / (1.0f + expf(-x))) : tanhf(x);
        Out[idx] = x;
      }
    }
  }
}

// ---------------------------------------------------------------------------
extern "C" void kernel_launch(void* const* d_in, const int* in_sizes, int n_in,
                              void* d_out, int out_size, void* d_ws, size_t ws_size,
                              hipStream_t stream) {
  const float* X  = (const float*)d_in[0];
  const int*   ei = (const int*)d_in[1];   // [2, E]: row0 = src, row1 = dst
  const float* H  = (const float*)d_in[2];
  const float* Wu = (const float*)d_in[3];
  const float* bu = (const float*)d_in[4];
  const float* Wr = (const float*)d_in[5];
  const float* br = (const float*)d_in[6];
  const float* Wc = (const float*)d_in[7];
  const float* bc = (const float*)d_in[8];
  const float* Lu = (const float*)d_in[9];
  const float* lub = (const float*)d_in[10];
  const float* Lr = (const float*)d_in[11];
  const float* lrb = (const float*)d_in[12];
  const float* Lc = (const float*)d_in[13];
  const float* lcb = (const float*)d_in[14];
  float* out = (float*)d_out;

  const int* src = ei;
  const int* dst = ei + N_EDGES;

  // workspace layout (floats): dinv[N] | Agg[NC] | Zu[NC] | Zr[NC] | Zc[NC]
  float* ws   = (float*)d_ws;
  float* dinv = ws;
  float* Agg  = ws + N_NODES;
  float* Zu   = Agg + NC;
  float* Zr   = Zu + NC;
  float* Zc   = Zr + NC;

  const int nb256 = (N_NODES + 255) / 256;

  // 1) degree -> dinv (cnt overlays dinv buffer, transformed in place)
  k_zero_i32<<<nb256, 256, 0, stream>>>((int*)dinv, N_NODES);
  k_count_deg<<<N_EDGES / 256, 256, 0, stream>>>(dst, (int*)dinv);
  k_make_dinv<<<nb256, 256, 0, stream>>>((const int*)dinv, dinv, N_NODES);

  // 2) Agg = Ahat @ X  (self-loop init + edge scatter-add)
  k_self_init<<<NC / 256, 256, 0, stream>>>(X, dinv, Agg);
  k_edge_agg<<<N_EDGES / 8, 256, 0, stream>>>(src, dst, X, dinv, Agg);

  // 3) Zk = Agg @ Wk + bk   (WMMA f32, 32 rows/block -> 3125 blocks)
  k_gemm_xw<<<N_NODES / 32, 256, 0, stream>>>(Agg, Wu, bu, Zu);
  k_gemm_xw<<<N_NODES / 32, 256, 0, stream>>>(Agg, Wr, br, Zr);
  k_gemm_xw<<<N_NODES / 32, 256, 0, stream>>>(Agg, Wc, bc, Zc);

  // 4) gates (in-place: U overwrites Zu, R overwrites Zr)
  k_gemm_gate<<<N_NODES / 32, 256, 0, stream>>>(Zu, H, nullptr, Lu, lub, Zu, 0,
                                                nullptr, nullptr);
  k_gemm_gate<<<N_NODES / 32, 256, 0, stream>>>(Zr, H, nullptr, Lr, lrb, Zr, 0,
                                                nullptr, nullptr);
  // candidate gate + fused final combine: out = U*H + (1-U)*tanh([Zc|H*R]@Lc+b)
  k_gemm_gate<<<N_NODES / 32, 256, 0, stream>>>(Zc, H, Zr, Lc, lcb, out, 1,
                                                Zu, H);
}